// GCNLayer_36532991819875
// MI455X (gfx1250) — compile-verified
//
#include <hip/hip_runtime.h>
#include <hip/hip_bf16.h>
#include <math.h>

typedef float v2f __attribute__((ext_vector_type(2)));
typedef float v4f __attribute__((ext_vector_type(4)));
typedef float v8f __attribute__((ext_vector_type(8)));

#define IN_DIM 128
#define OUT_DIM 64
#define NEG_SLOPE 0.01f

// ---------------------------------------------------------------------------
// Kernel 1: zero the output accumulator, init segment-max to -inf, denom to 0.
// (d_out / d_ws are poisoned by the harness and not re-poisoned between
//  replays, so we must fully re-initialize every call.)
// ---------------------------------------------------------------------------
__global__ void init_kernel(float* __restrict__ h_out, float* __restrict__ e_max,
                            float* __restrict__ denom, int N) {
  int idx = blockIdx.x * blockDim.x + threadIdx.x;
  if (idx < N * OUT_DIM) h_out[idx] = 0.0f;
  if (idx < N) {
    e_max[idx] = -INFINITY;
    denom[idx] = 0.0f;
  }
}

// ---------------------------------------------------------------------------
// Kernel 2: z = h @ W1^T via V_WMMA_F32_16X16X4_F32 (f32 matrix cores).
// One wave computes a 16(M) x 64(N) tile of z; K=128 is consumed 4 at a time.
//
// A (16x4 f32, 2 VGPRs): lanes 0-15 row M=lane, K = k0+{0,1};
//                        lanes 16-31 row M=lane-16, K = k0+{2,3}.
// B (4x16 f32, 2 VGPRs): lanes 0-15 col N=lane, K = k0+{0,1};
//                        lanes 16-31 col N=lane-16, K = k0+{2,3}.
//                        B[k][n] = W1[n][k]  (z = h @ W1^T).
// C/D (16x16 f32, 8 VGPRs): VGPR j -> row j (lanes 0-15) / row j+8 (16-31).
// ---------------------------------------------------------------------------
__global__ void gemm_z_kernel(const float* __restrict__ h,
                              const float* __restrict__ W1,
                              float* __restrict__ z, int N) {
  const int lane   = threadIdx.x & 31;
  const int lane16 = lane & 15;
  const int half   = lane >> 4;          // 0: lanes 0-15, 1: lanes 16-31
  const int wave   = (blockIdx.x * blockDim.x + threadIdx.x) >> 5;
  const int numTiles = (N + 15) >> 4;
  if (wave >= numTiles) return;          // wave-uniform: EXEC stays all-ones

  int row = wave * 16 + lane16;
  if (row >= N) row = N - 1;             // clamp loads (stores are guarded)

  const float* __restrict__ hrow = h + (long)row * IN_DIM;
  const float* __restrict__ w0 = W1 + (long)( 0 + lane16) * IN_DIM;
  const float* __restrict__ w1 = W1 + (long)(16 + lane16) * IN_DIM;
  const float* __restrict__ w2 = W1 + (long)(32 + lane16) * IN_DIM;
  const float* __restrict__ w3 = W1 + (long)(48 + lane16) * IN_DIM;

  v8f acc0 = {}, acc1 = {}, acc2 = {}, acc3 = {};

  for (int k0 = 0; k0 < IN_DIM; k0 += 4) {
    const int kk = k0 + 2 * half;
    v2f a  = *(const v2f*)(hrow + kk);
    v2f b0 = *(const v2f*)(w0 + kk);
    v2f b1 = *(const v2f*)(w1 + kk);
    v2f b2 = *(const v2f*)(w2 + kk);
    v2f b3 = *(const v2f*)(w3 + kk);
    acc0 = __builtin_amdgcn_wmma_f32_16x16x4_f32(false, a, false, b0, (short)0, acc0, false, false);
    acc1 = __builtin_amdgcn_wmma_f32_16x16x4_f32(false, a, false, b1, (short)0, acc1, false, false);
    acc2 = __builtin_amdgcn_wmma_f32_16x16x4_f32(false, a, false, b2, (short)0, acc2, false, false);
    acc3 = __builtin_amdgcn_wmma_f32_16x16x4_f32(false, a, false, b3, (short)0, acc3, false, false);
  }

  const int rbase = wave * 16 + half * 8;
  for (int j = 0; j < 8; ++j) {
    const int r = rbase + j;
    if (r < N) {
      float* __restrict__ zr = z + (long)r * OUT_DIM;
      zr[ 0 + lane16] = acc0[j];
      zr[16 + lane16] = acc1[j];
      zr[32 + lane16] = acc2[j];
      zr[48 + lane16] = acc3[j];
    }
  }
}

// ---------------------------------------------------------------------------
// Kernel 3: per-node attention scalars s_src[n] = z[n].a_src, s_dst = z[n].a_dst
// 16 lanes per node, float4 loads, shuffle reduction within the 16-lane group.
// This factors the per-edge 64-dim dot products into two 4-byte gathers.
// ---------------------------------------------------------------------------
__global__ void dots_kernel(const float* __restrict__ z, const float* __restrict__ Wa,
                            float* __restrict__ s_src, float* __restrict__ s_dst, int N) {
  const int tid = blockIdx.x * blockDim.x + threadIdx.x;
  const int node = tid >> 4;
  const int lane16 = tid & 15;
  if (node >= N) return;
  v4f zv = *(const v4f*)(z + (long)node * OUT_DIM + lane16 * 4);
  v4f as = *(const v4f*)(Wa + lane16 * 4);
  v4f ad = *(const v4f*)(Wa + OUT_DIM + lane16 * 4);
  float ps = zv.x * as.x + zv.y * as.y + zv.z * as.z + zv.w * as.w;
  float pd = zv.x * ad.x + zv.y * ad.y + zv.z * ad.z + zv.w * ad.w;
#pragma unroll
  for (int m = 8; m >= 1; m >>= 1) {     // stays inside the 16-lane group
    ps += __shfl_xor(ps, m, 32);
    pd += __shfl_xor(pd, m, 32);
  }
  if (lane16 == 0) {
    s_src[node] = ps;
    s_dst[node] = pd;
  }
}

// Float atomic-max via signed-max / unsigned-min bit trick (valid with -inf init).
__device__ __forceinline__ void atomicMaxF(float* addr, float val) {
  if (val >= 0.0f)
    atomicMax((int*)addr, __float_as_int(val));
  else
    atomicMin((unsigned int*)addr, __float_as_uint(val));
}

// ---------------------------------------------------------------------------
// Kernel 4: edge logits + leaky-relu + segment max over dst.
// ---------------------------------------------------------------------------
__global__ void edge_logit_kernel(const int* __restrict__ src, const int* __restrict__ dst,
                                  const float* __restrict__ s_src,
                                  const float* __restrict__ s_dst,
                                  float* __restrict__ e_buf, float* __restrict__ e_max,
                                  int E) {
  const int e = blockIdx.x * blockDim.x + threadIdx.x;
  if (e >= E) return;
  float v = s_src[src[e]] + s_dst[dst[e]];
  v = (v >= 0.0f) ? v : NEG_SLOPE * v;
  e_buf[e] = v;
  atomicMaxF(&e_max[dst[e]], v);
}

// Kernel 5: e_max = isfinite(e_max) ? e_max : 0   (nodes with no in-edges)
__global__ void fix_emax_kernel(float* __restrict__ e_max, int N) {
  const int i = blockIdx.x * blockDim.x + threadIdx.x;
  if (i >= N) return;
  float m = e_max[i];
  e_max[i] = __builtin_isfinite(m) ? m : 0.0f;
}

// ---------------------------------------------------------------------------
// Kernel 6: e_exp = exp(e - e_max[dst]) (stored in place) + segment sum denom.
// ---------------------------------------------------------------------------
__global__ void edge_exp_kernel(const int* __restrict__ dst, const float* __restrict__ e_max,
                                float* __restrict__ e_buf, float* __restrict__ denom, int E) {
  const int e = blockIdx.x * blockDim.x + threadIdx.x;
  if (e >= E) return;
  const int d = dst[e];
  float v = expf(e_buf[e] - e_max[d]);
  e_buf[e] = v;
  atomicAdd(&denom[d], v);
}

// ---------------------------------------------------------------------------
// Kernel 7: h_out[dst] += alpha * z[src].  16 lanes per edge, float4 gathers
// of z[src] (L2-resident: z is 25.6 MB < 192 MB L2) and f32 atomic adds into
// h_out[dst] (also L2-resident).
// ---------------------------------------------------------------------------
__global__ void aggregate_kernel(const int* __restrict__ src, const int* __restrict__ dst,
                                 const float* __restrict__ e_buf,
                                 const float* __restrict__ denom,
                                 const float* __restrict__ z,
                                 float* __restrict__ out, int E) {
  const int tid = blockIdx.x * blockDim.x + threadIdx.x;
  const int e = tid >> 4;
  const int lane16 = tid & 15;
  if (e >= E) return;
  const int s = src[e];
  const int d = dst[e];
  const float alpha = e_buf[e] / fmaxf(denom[d], 1e-16f);
  v4f zv = *(const v4f*)(z + (long)s * OUT_DIM + lane16 * 4);
  float* o = out + (long)d * OUT_DIM + lane16 * 4;
  atomicAdd(o + 0, alpha * zv.x);
  atomicAdd(o + 1, alpha * zv.y);
  atomicAdd(o + 2, alpha * zv.z);
  atomicAdd(o + 3, alpha * zv.w);
}

// ---------------------------------------------------------------------------
extern "C" void kernel_launch(void* const* d_in, const int* in_sizes, int n_in,
                              void* d_out, int out_size, void* d_ws, size_t ws_size,
                              hipStream_t stream) {
  const float* h  = (const float*)d_in[0];   // [N,128]
  const int*   src = (const int*)d_in[1];    // [E]
  const int*   dst = (const int*)d_in[2];    // [E]
  const float* W1 = (const float*)d_in[3];   // [64,128]
  const float* Wa = (const float*)d_in[4];   // [1,128]

  const int N = in_sizes[0] / IN_DIM;
  const int E = in_sizes[1];
  float* out = (float*)d_out;                // [N,64]

  // Workspace layout (floats): z[N*64] | s_src[N] | s_dst[N] | e_max[N] |
  //                            denom[N] | e_buf[E]   (~34 MB total)
  float* ws    = (float*)d_ws;
  float* z     = ws;
  float* s_src = z + (size_t)N * OUT_DIM;
  float* s_dst = s_src + N;
  float* e_max = s_dst + N;
  float* denom = e_max + N;
  float* e_buf = denom + N;

  const int B = 256;  // 8 wave32s per block
  init_kernel<<<(N * OUT_DIM + B - 1) / B, B, 0, stream>>>(out, e_max, denom, N);

  const int numTiles = (N + 15) / 16;
  gemm_z_kernel<<<(numTiles * 32 + B - 1) / B, B, 0, stream>>>(h, W1, z, N);

  dots_kernel<<<(N * 16 + B - 1) / B, B, 0, stream>>>(z, Wa, s_src, s_dst, N);

  edge_logit_kernel<<<(E + B - 1) / B, B, 0, stream>>>(src, dst, s_src, s_dst,
                                                       e_buf, e_max, E);
  fix_emax_kernel<<<(N + B - 1) / B, B, 0, stream>>>(e_max, N);
  edge_exp_kernel<<<(E + B - 1) / B, B, 0, stream>>>(dst, e_max, e_buf, denom, E);

  const long aggThreads = (long)E * 16;
  aggregate_kernel<<<(unsigned int)((aggThreads + B - 1) / B), B, 0, stream>>>(
      src, dst, e_buf, denom, z, out, E);
}